// ODEBlock_2929167696197
// MI455X (gfx1250) — compile-verified
//
#include <hip/hip_runtime.h>
#include <math.h>

typedef float v2f __attribute__((ext_vector_type(2)));
typedef float v8f __attribute__((ext_vector_type(8)));

#define NEL (1024 * 512)
#define NELF 524288.0f

// scalar slots in workspace
#define S_T      0
#define S_DT     1
#define S_DONE   2
#define S_COMMIT 3
#define S_H0     4
#define S_D1     5

// ---------------------------------------------------------------------------
// GEMM: C[M,N] = act(A[M,K] @ W[K,N] + bias), fp32 WMMA 16x16x4.
// Block = 128 threads = 4 waves (2x2), block tile 64x64, K staged 32 in LDS,
// double-buffered with register prefetch so global-load latency overlaps WMMA.
// ---------------------------------------------------------------------------
#define GM 64
#define GN 64
#define GK 32
#define LDA 36   // padded LDS row stride (floats): conflict-free frag reads, 16B aligned
#define LDB 68

__global__ __launch_bounds__(128)
void gemm_kernel(const float* __restrict__ A, const float* __restrict__ W,
                 const float* __restrict__ bias, float* __restrict__ C,
                 int M, int N, int K, int do_relu)
{
    __shared__ float As[2][GM * LDA];
    __shared__ float Bs[2][GK * LDB];

    const int tid  = threadIdx.x;
    const int lane = tid & 31;
    const int wave = tid >> 5;     // 0..3
    const int wm   = wave >> 1;    // 0..1 (M dir)
    const int wn   = wave & 1;     // 0..1 (N dir)
    const int half = lane >> 4;    // 0..1
    const int lr   = lane & 15;

    const int mbase = blockIdx.y * GM;
    const int nbase = blockIdx.x * GN;

    v8f acc[2][2];
#pragma unroll
    for (int a = 0; a < 2; ++a)
#pragma unroll
        for (int b = 0; b < 2; ++b)
            acc[a][b] = (v8f){0.f, 0.f, 0.f, 0.f, 0.f, 0.f, 0.f, 0.f};

    float4 ra[4], rb[4];

    // ---- prologue: fetch chunk 0 and stage into LDS buffer 0 ----
#pragma unroll
    for (int i = 0; i < 4; ++i) {
        int idx4 = tid + i * 128;
        int rA = idx4 >> 3, cA = idx4 & 7;        // A tile 64x32
        ra[i] = *(const float4*)(A + (size_t)(mbase + rA) * K + cA * 4);
        int rB = idx4 >> 4, cB = idx4 & 15;       // B tile 32x64
        rb[i] = *(const float4*)(W + (size_t)rB * N + nbase + cB * 4);
    }
#pragma unroll
    for (int i = 0; i < 4; ++i) {
        int idx4 = tid + i * 128;
        int rA = idx4 >> 3, cA = idx4 & 7;
        *(float4*)&As[0][rA * LDA + cA * 4] = ra[i];
        int rB = idx4 >> 4, cB = idx4 & 15;
        *(float4*)&Bs[0][rB * LDB + cB * 4] = rb[i];
    }
    __syncthreads();

    const int NC = K / GK;   // 16 chunks
    for (int c = 0; c < NC; ++c) {
        const int cur = c & 1;
        const int nxt = cur ^ 1;

        // issue next chunk's global loads BEFORE computing (latency overlap)
        if (c + 1 < NC) {
            const int k0 = (c + 1) * GK;
#pragma unroll
            for (int i = 0; i < 4; ++i) {
                int idx4 = tid + i * 128;
                int rA = idx4 >> 3, cA = idx4 & 7;
                ra[i] = *(const float4*)(A + (size_t)(mbase + rA) * K + k0 + cA * 4);
                int rB = idx4 >> 4, cB = idx4 & 15;
                rb[i] = *(const float4*)(W + (size_t)(k0 + rB) * N + nbase + cB * 4);
            }
        }

        // compute on current buffer: 8 K-steps x 4 wmma
#pragma unroll
        for (int kk = 0; kk < GK; kk += 4) {
            const int ks = kk + 2 * half;   // per-lane K pair per ISA A/B layout
            v2f a0, a1, b0, b1;
            a0.x = As[cur][(wm * 32 +      lr) * LDA + ks];
            a0.y = As[cur][(wm * 32 +      lr) * LDA + ks + 1];
            a1.x = As[cur][(wm * 32 + 16 + lr) * LDA + ks];
            a1.y = As[cur][(wm * 32 + 16 + lr) * LDA + ks + 1];
            b0.x = Bs[cur][ ks      * LDB + wn * 32 +      lr];
            b0.y = Bs[cur][(ks + 1) * LDB + wn * 32 +      lr];
            b1.x = Bs[cur][ ks      * LDB + wn * 32 + 16 + lr];
            b1.y = Bs[cur][(ks + 1) * LDB + wn * 32 + 16 + lr];
            acc[0][0] = __builtin_amdgcn_wmma_f32_16x16x4_f32(false, a0, false, b0, (short)0, acc[0][0], false, false);
            acc[0][1] = __builtin_amdgcn_wmma_f32_16x16x4_f32(false, a0, false, b1, (short)0, acc[0][1], false, false);
            acc[1][0] = __builtin_amdgcn_wmma_f32_16x16x4_f32(false, a1, false, b0, (short)0, acc[1][0], false, false);
            acc[1][1] = __builtin_amdgcn_wmma_f32_16x16x4_f32(false, a1, false, b1, (short)0, acc[1][1], false, false);
        }

        // stage next chunk into the other buffer; single barrier per chunk
        if (c + 1 < NC) {
#pragma unroll
            for (int i = 0; i < 4; ++i) {
                int idx4 = tid + i * 128;
                int rA = idx4 >> 3, cA = idx4 & 7;
                *(float4*)&As[nxt][rA * LDA + cA * 4] = ra[i];
                int rB = idx4 >> 4, cB = idx4 & 15;
                *(float4*)&Bs[nxt][rB * LDB + cB * 4] = rb[i];
            }
            __syncthreads();
        }
    }

    // epilogue: bias (+ ReLU) and store; C VGPRv -> M = v + 8*half
#pragma unroll
    for (int ms = 0; ms < 2; ++ms) {
#pragma unroll
        for (int ns = 0; ns < 2; ++ns) {
            int col = nbase + wn * 32 + ns * 16 + lr;
            float bv = bias[col];
#pragma unroll
            for (int v = 0; v < 8; ++v) {
                int row = mbase + wm * 32 + ms * 16 + v + 8 * half;
                float x = acc[ms][ns][v] + bv;
                if (do_relu) x = fmaxf(x, 0.f);
                C[(size_t)row * N + col] = x;
            }
        }
    }
}

// ---------------------------------------------------------------------------
// Elementwise / control kernels
// ---------------------------------------------------------------------------
__global__ void init_y0_kernel(const float* __restrict__ x, float* __restrict__ y)
{
    int i = blockIdx.x * blockDim.x + threadIdx.x;   // 0..NEL-1
    int r = i >> 9, c = i & 511;
    y[i] = (c < 448) ? x[r * 448 + c] : 0.f;
}

// out = y + s * (c1*p1 + ... + c6*p6);  s = min(dt, 1-t) (mode 0) or h0 (mode 1)
__global__ void combine_kernel(float* __restrict__ out, const float* __restrict__ y,
                               const float* __restrict__ p1, const float* __restrict__ p2,
                               const float* __restrict__ p3, const float* __restrict__ p4,
                               const float* __restrict__ p5, const float* __restrict__ p6,
                               float c1, float c2, float c3, float c4, float c5, float c6,
                               const float* __restrict__ scal, int scale_mode)
{
    int i = (blockIdx.x * blockDim.x + threadIdx.x) * 4;
    float s = (scale_mode == 0) ? fminf(scal[S_DT], 1.f - scal[S_T]) : scal[S_H0];
    float4 acc = {0.f, 0.f, 0.f, 0.f};
#define ACCUM(P, C) do { if ((C) != 0.f) { float4 v = *(const float4*)((P) + i); \
        acc.x += (C) * v.x; acc.y += (C) * v.y; acc.z += (C) * v.z; acc.w += (C) * v.w; } } while (0)
    ACCUM(p1, c1); ACCUM(p2, c2); ACCUM(p3, c3);
    ACCUM(p4, c4); ACCUM(p5, c5); ACCUM(p6, c6);
#undef ACCUM
    float4 yy = *(const float4*)(y + i);
    float4 o;
    o.x = yy.x + s * acc.x; o.y = yy.y + s * acc.y;
    o.z = yy.z + s * acc.z; o.w = yy.w + s * acc.w;
    *(float4*)(out + i) = o;
}

// deterministic two-pass reduction, pass 1: 256 blocks x 256 threads x 8 elems
__global__ void reduce_kernel(const float* __restrict__ y, const float* __restrict__ y5,
                              const float* __restrict__ f0, const float* __restrict__ f1,
                              const float* __restrict__ k1, const float* __restrict__ k3,
                              const float* __restrict__ k4, const float* __restrict__ k5,
                              const float* __restrict__ k6, const float* __restrict__ k7,
                              const float* __restrict__ scal, float* __restrict__ partial, int mode)
{
    __shared__ float sm[256];
    const float E1 = (float)( 71.0 /  57600.0);
    const float E3 = (float)(-71.0 /  16695.0);
    const float E4 = (float)( 71.0 /   1920.0);
    const float E5 = (float)(-17253.0 / 339200.0);
    const float E6 = (float)( 22.0 /    525.0);
    const float E7 = (float)(-1.0 /      40.0);
    int t = threadIdx.x, b = blockIdx.x;
    float s = 0.f;
    for (int j = 0; j < 8; ++j) {
        int i = b * 256 + t + j * 65536;
        float v;
        if (mode == 0) {
            float sc = 1e-3f + 1e-3f * fabsf(y[i]);
            v = y[i] / sc;
        } else if (mode == 1) {
            float sc = 1e-3f + 1e-3f * fabsf(y[i]);
            v = f0[i] / sc;
        } else if (mode == 2) {
            float sc = 1e-3f + 1e-3f * fabsf(y[i]);
            v = (f1[i] - f0[i]) / sc;
        } else {
            float dtc = fminf(scal[S_DT], 1.f - scal[S_T]);
            float err = dtc * (E1 * k1[i] + E3 * k3[i] + E4 * k4[i] +
                               E5 * k5[i] + E6 * k6[i] + E7 * k7[i]);
            float sc = 1e-3f + 1e-3f * fmaxf(fabsf(y[i]), fabsf(y5[i]));
            v = err / sc;
        }
        s += v * v;
    }
    sm[t] = s; __syncthreads();
    for (int o = 128; o > 0; o >>= 1) { if (t < o) sm[t] += sm[t + o]; __syncthreads(); }
    if (t == 0) partial[b] = sm[0];
}

__global__ void finalize_h0_kernel(const float* __restrict__ pA, const float* __restrict__ pB,
                                   float* __restrict__ scal)
{
    __shared__ float sa[256], sb[256];
    int t = threadIdx.x;
    sa[t] = pA[t]; sb[t] = pB[t]; __syncthreads();
    for (int o = 128; o > 0; o >>= 1) { if (t < o) { sa[t] += sa[t + o]; sb[t] += sb[t + o]; } __syncthreads(); }
    if (t == 0) {
        float d0 = sqrtf(sa[0] / NELF);
        float d1 = sqrtf(sb[0] / NELF);
        float h0 = (fminf(d0, d1) < 1e-5f) ? 1e-6f : 0.01f * d0 / fmaxf(d1, 1e-12f);
        scal[S_H0] = h0;
        scal[S_D1] = d1;
    }
}

__global__ void finalize_dt0_kernel(const float* __restrict__ pA, float* __restrict__ scal)
{
    __shared__ float sa[256];
    int t = threadIdx.x;
    sa[t] = pA[t]; __syncthreads();
    for (int o = 128; o > 0; o >>= 1) { if (t < o) sa[t] += sa[t + o]; __syncthreads(); }
    if (t == 0) {
        float h0 = scal[S_H0], d1 = scal[S_D1];
        float d2 = sqrtf(sa[0] / NELF) / h0;
        float dmax = fmaxf(d1, d2);
        float h1 = (dmax <= 1e-15f) ? fmaxf(1e-6f, h0 * 1e-3f)
                                    : powf(0.01f / fmaxf(dmax, 1e-12f), 0.2f);
        float dt0 = fminf(fminf(100.f * h0, h1), 1.0f);
        scal[S_T] = 0.f; scal[S_DT] = dt0; scal[S_DONE] = 0.f; scal[S_COMMIT] = 0.f;
    }
}

__global__ void finalize_step_kernel(const float* __restrict__ pA, float* __restrict__ scal)
{
    __shared__ float sa[256];
    int t = threadIdx.x;
    sa[t] = pA[t]; __syncthreads();
    for (int o = 128; o > 0; o >>= 1) { if (t < o) sa[t] += sa[t + o]; __syncthreads(); }
    if (t == 0) {
        float t0 = scal[S_T], dt = scal[S_DT], done = scal[S_DONE];
        float dtc = fminf(dt, 1.f - t0);
        float errn = sqrtf(sa[0] / NELF);
        int accept = (errn <= 1.0f) ? 1 : 0;
        float en  = fmaxf(errn, 1e-10f);
        float fac = fminf(fmaxf(0.9f * powf(en, -0.2f), 0.2f), 10.0f);
        float dtn = dtc * fac;
        int dprev = (done > 0.5f) ? 1 : 0;
        scal[S_COMMIT] = (!dprev && accept) ? 1.f : 0.f;
        if (!dprev) {
            if (accept) {
                scal[S_T] = t0 + dtc;
                if (t0 + dtc >= 1.0f - 1e-7f) scal[S_DONE] = 1.f;
            }
            scal[S_DT] = dtn;
        }
    }
}

__global__ void commit_kernel(float* __restrict__ y, float* __restrict__ F,
                              const float* __restrict__ y5, const float* __restrict__ k7,
                              const float* __restrict__ scal)
{
    if (scal[S_COMMIT] < 0.5f) return;
    int i = (blockIdx.x * blockDim.x + threadIdx.x) * 4;
    *(float4*)(y + i) = *(const float4*)(y5 + i);
    *(float4*)(F + i) = *(const float4*)(k7 + i);
}

__global__ void copy_kernel(float* __restrict__ out, const float* __restrict__ y)
{
    int i = (blockIdx.x * blockDim.x + threadIdx.x) * 4;
    *(float4*)(out + i) = *(const float4*)(y + i);
}

// ---------------------------------------------------------------------------
extern "C" void kernel_launch(void* const* d_in, const int* in_sizes, int n_in,
                              void* d_out, int out_size, void* d_ws, size_t ws_size,
                              hipStream_t stream)
{
    (void)in_sizes; (void)n_in; (void)out_size; (void)ws_size;
    const float* x  = (const float*)d_in[0];
    const float* W1 = (const float*)d_in[1];
    const float* b1 = (const float*)d_in[2];
    const float* W2 = (const float*)d_in[3];
    const float* b2 = (const float*)d_in[4];
    const float* W3 = (const float*)d_in[5];
    const float* b3 = (const float*)d_in[6];

    float* ws   = (float*)d_ws;
    float* scal = ws;
    float* pA   = ws + 256;
    float* pB   = ws + 512;
    float* base = ws + 1024;
    float* y   = base + (size_t)0  * NEL;
    float* F   = base + (size_t)1  * NEL;   // k1 = f(y) (FSAL carry)
    float* k2  = base + (size_t)2  * NEL;
    float* k3  = base + (size_t)3  * NEL;
    float* k4  = base + (size_t)4  * NEL;
    float* k5  = base + (size_t)5  * NEL;
    float* k6  = base + (size_t)6  * NEL;
    float* k7  = base + (size_t)7  * NEL;
    float* yst = base + (size_t)8  * NEL;
    float* y5v = base + (size_t)9  * NEL;
    float* h1  = base + (size_t)10 * NEL;
    float* h2  = base + (size_t)11 * NEL;

    const dim3 gGemm(512 / GN, 1024 / GM);  // (8,16)

    auto feval = [&](const float* yin, float* kout) {
        gemm_kernel<<<gGemm, 128, 0, stream>>>(yin, W1, b1, h1, 1024, 512, 512, 1);
        gemm_kernel<<<gGemm, 128, 0, stream>>>(h1,  W2, b2, h2, 1024, 512, 512, 1);
        gemm_kernel<<<gGemm, 128, 0, stream>>>(h2,  W3, b3, kout, 1024, 512, 512, 0);
    };
    auto combine = [&](float* out, float c1, float c2, float c3, float c4, float c5, float c6, int mode) {
        combine_kernel<<<512, 256, 0, stream>>>(out, y, F, k2, k3, k4, k5, k6,
                                                c1, c2, c3, c4, c5, c6, scal, mode);
    };

    // dopri5 tableau (match f64->f32 rounding of the reference constants)
    const float A21 = (float)(1.0/5.0);
    const float A31 = (float)(3.0/40.0),      A32 = (float)(9.0/40.0);
    const float A41 = (float)(44.0/45.0),     A42 = (float)(-56.0/15.0),    A43 = (float)(32.0/9.0);
    const float A51 = (float)(19372.0/6561.0),A52 = (float)(-25360.0/2187.0),
                A53 = (float)(64448.0/6561.0),A54 = (float)(-212.0/729.0);
    const float A61 = (float)(9017.0/3168.0), A62 = (float)(-355.0/33.0),
                A63 = (float)(46732.0/5247.0),A64 = (float)(49.0/176.0),    A65 = (float)(-5103.0/18656.0);
    const float B1  = (float)(35.0/384.0),    B3  = (float)(500.0/1113.0),
                B4  = (float)(125.0/192.0),   B5  = (float)(-2187.0/6784.0),B6  = (float)(11.0/84.0);

    // ---- init: y0, f0, Hairer initial step size ----
    init_y0_kernel<<<2048, 256, 0, stream>>>(x, y);
    feval(y, F);                                                                  // f0
    reduce_kernel<<<256, 256, 0, stream>>>(y, y, F, F, F, F, F, F, F, F, scal, pA, 0); // d0
    reduce_kernel<<<256, 256, 0, stream>>>(y, y, F, F, F, F, F, F, F, F, scal, pB, 1); // d1
    finalize_h0_kernel<<<1, 256, 0, stream>>>(pA, pB, scal);
    combine(yst, 1.f, 0.f, 0.f, 0.f, 0.f, 0.f, 1);                                // y + h0*f0
    feval(yst, k2);                                                               // f1 (scratch in k2)
    reduce_kernel<<<256, 256, 0, stream>>>(y, y, F, k2, F, F, F, F, F, F, scal, pA, 2); // d2
    finalize_dt0_kernel<<<1, 256, 0, stream>>>(pA, scal);

    // ---- 20 fixed dopri5 steps (data-gated, like lax.scan) ----
    for (int s = 0; s < 20; ++s) {
        combine(yst, A21, 0.f, 0.f, 0.f, 0.f, 0.f, 0); feval(yst, k2);
        combine(yst, A31, A32, 0.f, 0.f, 0.f, 0.f, 0); feval(yst, k3);
        combine(yst, A41, A42, A43, 0.f, 0.f, 0.f, 0); feval(yst, k4);
        combine(yst, A51, A52, A53, A54, 0.f, 0.f, 0); feval(yst, k5);
        combine(yst, A61, A62, A63, A64, A65, 0.f, 0); feval(yst, k6);
        combine(y5v, B1, 0.f, B3, B4, B5, B6, 0);                                 // 5th order solution
        feval(y5v, k7);                                                           // FSAL stage
        reduce_kernel<<<256, 256, 0, stream>>>(y, y5v, F, F, F, k3, k4, k5, k6, k7, scal, pA, 3);
        finalize_step_kernel<<<1, 256, 0, stream>>>(pA, scal);
        commit_kernel<<<512, 256, 0, stream>>>(y, F, y5v, k7, scal);
    }

    copy_kernel<<<512, 256, 0, stream>>>((float*)d_out, y);
}